// GaussianRasterizer_11441792877129
// MI455X (gfx1250) — compile-verified
//
#include <hip/hip_runtime.h>
#include <hip/hip_bf16.h>
#include <stdint.h>

#define W_IMG 128
#define H_IMG 128
#define N_GS 1024
#define TANX 0.5f
#define TANY 0.5f
#define LOWPASS 0.3f
#define MIN_DEPTH 0.2f
#define MIN_ALPHA (1.0f/255.0f)
#define T_EPS 1e-4f
#define REC_STRIDE 12            // floats per gaussian record (48 B)
#define REC_FLOATS (N_GS * REC_STRIDE)
#define REC_CHUNK16 (REC_FLOATS / 4)   // number of 16-byte chunks = 3072

// record layout (48 B): [0]=conA [1]=conB [2]=conC [3]=pix_x [4]=pix_y
//                       [5..7]=pad  [8]=op_eff [9]=r [10]=g [11]=b
// -> conic block is b128@0 + b32@16; op/color block is an aligned float4 @32.

typedef __attribute__((ext_vector_type(2))) float v2f;
typedef __attribute__((ext_vector_type(8))) float v8f;
typedef __attribute__((ext_vector_type(4))) int   v4i;

#if __has_builtin(__builtin_amdgcn_wmma_f32_16x16x4_f32)
#define HAVE_WMMA4 1
#endif
#if __has_builtin(__builtin_amdgcn_global_load_async_to_lds_b128)
#define HAVE_ASYNC 1
#endif

// generic -> addrspace casts via integer round-trip:
//  - global: flat address == global address
//  - LDS: flat LDS address carries the LDS byte offset in its low 32 bits (ISA 10.2)
typedef __attribute__((address_space(1))) v4i* gptr_v4i;
typedef __attribute__((address_space(3))) v4i* lptr_v4i;
#define AS1_V4I(p) ((gptr_v4i)(uintptr_t)(p))
#define AS3_V4I(p) ((lptr_v4i)(uint32_t)(uintptr_t)(p))

__device__ __forceinline__ void wave_lds_wait() {
  // wave-local wait for our own DS stores before cross-lane LDS reads
  asm volatile("s_wait_dscnt 0" ::: "memory");
}
__device__ __forceinline__ void wait_async_done() {
  asm volatile("s_wait_asynccnt 0" ::: "memory");
}

// -------------------------------------------------------------------------
// Kernel 1: per-gaussian preprocessing + depth bitonic sort + sorted records
// -------------------------------------------------------------------------
__global__ __launch_bounds__(1024) void gs_prep_sort(
    const float* __restrict__ means3D, const float* __restrict__ opac,
    const float* __restrict__ colors,  const float* __restrict__ scales,
    const float* __restrict__ rots,    const float* __restrict__ viewm,
    const float* __restrict__ projm,   float* __restrict__ recs)
{
  __shared__ __align__(16) float s_rec[REC_FLOATS];   // 48 KB
  __shared__ float s_key[N_GS];                       // 4 KB
  __shared__ int   s_idx[N_GS];                       // 4 KB

  const int i = threadIdx.x;

  const float mx = means3D[i*3+0], my = means3D[i*3+1], mz = means3D[i*3+2];

  // quaternion -> rotation
  float qr = rots[i*4+0], qx = rots[i*4+1], qy = rots[i*4+2], qz = rots[i*4+3];
  const float qn = rsqrtf(qr*qr + qx*qx + qy*qy + qz*qz);
  qr *= qn; qx *= qn; qy *= qn; qz *= qn;
  const float R00 = 1.f-2.f*(qy*qy+qz*qz), R01 = 2.f*(qx*qy-qr*qz), R02 = 2.f*(qx*qz+qr*qy);
  const float R10 = 2.f*(qx*qy+qr*qz), R11 = 1.f-2.f*(qx*qx+qz*qz), R12 = 2.f*(qy*qz-qr*qx);
  const float R20 = 2.f*(qx*qz-qr*qy), R21 = 2.f*(qy*qz+qr*qx), R22 = 1.f-2.f*(qx*qx+qy*qy);

  const float s0 = scales[i*3+0], s1 = scales[i*3+1], s2 = scales[i*3+2];
  // M = R * diag(s); cov3d = M M^T (symmetric)
  const float M00=R00*s0, M01=R01*s1, M02=R02*s2;
  const float M10=R10*s0, M11=R11*s1, M12=R12*s2;
  const float M20=R20*s0, M21=R21*s1, M22=R22*s2;
  const float c00 = M00*M00+M01*M01+M02*M02;
  const float c01 = M00*M10+M01*M11+M02*M12;
  const float c02 = M00*M20+M01*M21+M02*M22;
  const float c11 = M10*M10+M11*M11+M12*M12;
  const float c12 = M10*M20+M11*M21+M12*M22;
  const float c22 = M20*M20+M21*M21+M22*M22;

  // view transform
  const float W00=viewm[0], W01=viewm[1], W02=viewm[2],  V03=viewm[3];
  const float W10=viewm[4], W11=viewm[5], W12=viewm[6],  V13=viewm[7];
  const float W20=viewm[8], W21=viewm[9], W22=viewm[10], V23=viewm[11];
  const float tx = W00*mx + W01*my + W02*mz + V03;
  const float ty = W10*mx + W11*my + W12*mz + V13;
  const float tz = W20*mx + W21*my + W22*mz + V23;

  const float invz = 1.f / tz;
  const float limx = 1.3f * TANX, limy = 1.3f * TANY;
  const float txz = fminf(fmaxf(tx*invz, -limx), limx) * tz;
  const float tyz = fminf(fmaxf(ty*invz, -limy), limy) * tz;
  const float fx = (float)W_IMG / (2.f*TANX), fy = (float)H_IMG / (2.f*TANY);
  const float J00 = fx*invz, J02 = -fx*txz*invz*invz;
  const float J11 = fy*invz, J12 = -fy*tyz*invz*invz;
  // T2 = J @ W
  const float t00 = J00*W00 + J02*W20, t01 = J00*W01 + J02*W21, t02 = J00*W02 + J02*W22;
  const float t10 = J11*W10 + J12*W20, t11 = J11*W11 + J12*W21, t12 = J11*W12 + J12*W22;
  // cov2d = T2 C T2^T
  const float u0 = c00*t00 + c01*t01 + c02*t02;
  const float u1 = c01*t00 + c11*t01 + c12*t02;
  const float u2 = c02*t00 + c12*t01 + c22*t02;
  const float v00 = t00*u0 + t01*u1 + t02*u2;
  const float v01 = t10*u0 + t11*u1 + t12*u2;
  const float w0 = c00*t10 + c01*t11 + c02*t12;
  const float w1 = c01*t10 + c11*t11 + c12*t12;
  const float w2 = c02*t10 + c12*t11 + c22*t12;
  const float v11 = t10*w0 + t11*w1 + t12*w2;

  const float aa = v00 + LOWPASS, bb = v01, cc = v11 + LOWPASS;
  const float det = aa*cc - bb*bb;
  const float inv_det = (det != 0.f) ? (1.f/det) : 0.f;
  const float conA = cc*inv_det, conB = -bb*inv_det, conC = aa*inv_det;

  // projection -> pixel center
  const float ph0 = projm[0]*mx + projm[1]*my + projm[2]*mz + projm[3];
  const float ph1 = projm[4]*mx + projm[5]*my + projm[6]*mz + projm[7];
  const float ph3 = projm[12]*mx + projm[13]*my + projm[14]*mz + projm[15];
  const float pw  = 1.f / (ph3 + 1e-7f);
  const float pix_x = ((ph0*pw + 1.f)*(float)W_IMG - 1.f)*0.5f;
  const float pix_y = ((ph1*pw + 1.f)*(float)H_IMG - 1.f)*0.5f;

  const bool valid = (tz >= MIN_DEPTH) && (det > 0.f);
  const float op_eff = valid ? opac[i] : 0.f;   // invalid => alpha contribution exactly 0

  float* r = &s_rec[i*REC_STRIDE];
  r[0]=conA; r[1]=conB; r[2]=conC; r[3]=pix_x; r[4]=pix_y;
  r[5]=0.f; r[6]=0.f; r[7]=0.f;
  r[8]=op_eff; r[9]=colors[i*3+0]; r[10]=colors[i*3+1]; r[11]=colors[i*3+2];

  s_key[i] = tz; s_idx[i] = i;

  // bitonic sort (key=tz ascending, payload=index)
  for (int k = 2; k <= N_GS; k <<= 1) {
    for (int j = k >> 1; j > 0; j >>= 1) {
      __syncthreads();
      const int p = i ^ j;
      if (p > i) {
        const float ka = s_key[i], kb = s_key[p];
        const bool up = ((i & k) == 0);
        if ((ka > kb) == up) {
          const int ia = s_idx[i], ib = s_idx[p];
          s_key[i] = kb; s_key[p] = ka;
          s_idx[i] = ib; s_idx[p] = ia;
        }
      }
    }
  }
  __syncthreads();

  // gather sorted records to workspace (3x float4 per record)
  const int src = s_idx[i];
  #pragma unroll
  for (int w = 0; w < 3; ++w)
    ((float4*)recs)[i*3 + w] = ((const float4*)&s_rec[src*REC_STRIDE])[w];
}

// -------------------------------------------------------------------------
// Kernel 2: fused raster. Block = 32x8 pixel tile (256 threads, 8 waves).
// Wave = 8x4 pixel subtile, one pixel per lane. Per 16-gaussian chunk:
// power(u,v) = g . [1,u,v,u^2,v^2,uv]  via 2x V_WMMA_F32_16X16X4_F32 per
// 16-pixel tile (2 tiles/wave), transposed through wave-local LDS scratch,
// then sequential front-to-back alpha blending with early exit.
// -------------------------------------------------------------------------
#if defined(HAVE_WMMA4)
__device__ __forceinline__ v8f wmma4(v2f a, v2f b, v8f c) {
  return __builtin_amdgcn_wmma_f32_16x16x4_f32(false, a, false, b, (short)0, c, false, false);
}
#endif

__global__ __launch_bounds__(256) void gs_raster(
    const float* __restrict__ recs, const float* __restrict__ bgp,
    float* __restrict__ out)
{
  __shared__ __align__(16) float s_rec[REC_FLOATS];  // 48 KB: all sorted gaussians
  __shared__ __align__(16) float s_sc[8*32*16];      // 16 KB: per-wave power scratch

  const int tid = threadIdx.x;

  // ---- stage the whole sorted gaussian stream into LDS (async path) ----
#if defined(HAVE_ASYNC)
  {
    const char* g = (const char*)recs;
    char* l = (char*)s_rec;
    for (int k = tid; k < REC_CHUNK16; k += 256) {
      __builtin_amdgcn_global_load_async_to_lds_b128(
          AS1_V4I(g + (size_t)k*16), AS3_V4I(l + (size_t)k*16), 0, 0);
    }
    wait_async_done();
  }
#else
  for (int k = tid; k < REC_CHUNK16; k += 256)
    ((float4*)s_rec)[k] = ((const float4*)recs)[k];
#endif
  __syncthreads();

  const int lane = tid & 31, wv = tid >> 5;
  const int tileX = (blockIdx.x & 3) * 32;
  const int tileY = (blockIdx.x >> 2) * 8;
  const int sx = tileX + (wv & 3) * 8;     // wave subtile origin (8 wide x 4 tall)
  const int sy = tileY + (wv >> 2) * 4;
  const int pxi = sx + (lane & 7);
  const int pyi = sy + (lane >> 3);
  const float cxw = (float)sx + 3.5f;      // tile center for numerically-safe expansion
  const float cyw = (float)sy + 1.5f;

  const int Mrow = lane & 15;
  const bool lo = (lane < 16);             // k-half selector for A/B striping

#if defined(HAVE_WMMA4)
  // A-matrix features are constant per lane: f = [1,u,v,u^2,v^2,uv]
  v2f A1[2], A2[2];
  #pragma unroll
  for (int t = 0; t < 2; ++t) {
    const int p = Mrow + 16*t;
    const float u = (float)(p & 7) - 3.5f;
    const float v = (float)(p >> 3) - 1.5f;
    A1[t].x = lo ? 1.f : v;
    A1[t].y = lo ? u   : u*u;
    A2[t].x = lo ? v*v : 0.f;
    A2[t].y = lo ? u*v : 0.f;
  }
#endif

  float T = 1.f, cr = 0.f, cg = 0.f, cb = 0.f;
  bool done = false;
  const int scb = wv * 512;

  for (int c = 0; c < N_GS/16; ++c) {
#if defined(HAVE_WMMA4)
    {
      // build per-gaussian coefficient vector g for gaussian n = lane%16
      const float* rp = &s_rec[(c*16 + Mrow) * REC_STRIDE];
      const float4 con = *(const float4*)rp;          // conA conB conC pix_x
      const float py_g = rp[4];
      const float CA = con.x, CB = con.y, CC = con.z;
      const float dxg = con.w - cxw, dyg = py_g - cyw;
      const float g0 = -0.5f*(CA*dxg*dxg + CC*dyg*dyg) - CB*dxg*dyg;
      const float g1 = CA*dxg + CB*dyg;
      const float g2 = CC*dyg + CB*dxg;
      const float g3 = -0.5f*CA, g4 = -0.5f*CC, g5 = -CB;
      v2f B1, B2;
      B1.x = lo ? g0 : g2;  B1.y = lo ? g1 : g3;
      B2.x = lo ? g4 : 0.f; B2.y = lo ? g5 : 0.f;

      #pragma unroll
      for (int t = 0; t < 2; ++t) {
        v8f D = {};
        D = wmma4(A1[t], B1, D);     // K = 0..3
        D = wmma4(A2[t], B2, D);     // K = 4..7 (features 6,7 are zero padding)
        const int mb = 8 * (lane >> 4);
        #pragma unroll
        for (int r = 0; r < 8; ++r) {
          const int p = r + mb + 16*t;                  // strip pixel 0..31
          s_sc[scb + p*16 + Mrow] = D[r];               // transpose via LDS
        }
      }
      wave_lds_wait();   // our DS stores visible before cross-lane reads
    }
#endif
    // ---- sequential front-to-back blend for this lane's pixel ----
#if defined(HAVE_WMMA4)
    // batch-read all 16 powers for this pixel: 4x ds_load_b128, one wait
    const int base = scb + lane*16;
    float4 P[4];
    #pragma unroll
    for (int q = 0; q < 4; ++q)
      P[q] = *(const float4*)&s_sc[base + q*4];
#endif
    #pragma unroll
    for (int q = 0; q < 4; ++q) {
      #pragma unroll
      for (int m = 0; m < 4; ++m) {
        const int n = q*4 + m;
        const int gi = c*16 + n;
        // [op, r, g, b] as one aligned ds_load_b128
        const float4 cd = *(const float4*)&s_rec[gi*REC_STRIDE + 8];
#if defined(HAVE_WMMA4)
        const float pw = (&P[q].x)[m];
#else
        const float* rp2 = &s_rec[gi*REC_STRIDE];
        const float dx = rp2[3] - (float)pxi, dy = rp2[4] - (float)pyi;
        const float pw = -0.5f*(rp2[0]*dx*dx + rp2[2]*dy*dy) - rp2[1]*dx*dy;
#endif
        float a = 0.f;
        if (pw <= 0.f) a = fminf(0.99f, cd.x * __expf(pw));
        if (a < MIN_ALPHA) a = 0.f;
        if (!done && a > 0.f) {
          const float Tn = T * (1.f - a);
          if (Tn < T_EPS) {
            done = true;             // Tinc monotone: everything after is masked
          } else {
            const float wgt = a * T;
            cr += wgt * cd.y;
            cg += wgt * cd.z;
            cb += wgt * cd.w;
            T = Tn;
          }
        }
      }
    }
#if defined(HAVE_WMMA4)
    wave_lds_wait();   // power reads retired before next chunk's scatter
#endif
    if (__all(done ? 1 : 0)) break;  // wave-uniform: EXEC stays full for WMMA
  }

  const float bg0 = bgp[0], bg1 = bgp[1], bg2 = bgp[2];
  const int pix = pyi * W_IMG + pxi;
  out[0*W_IMG*H_IMG + pix] = cr + T * bg0;
  out[1*W_IMG*H_IMG + pix] = cg + T * bg1;
  out[2*W_IMG*H_IMG + pix] = cb + T * bg2;
  out[3*W_IMG*H_IMG + pix] = 1.f - T;
}

// -------------------------------------------------------------------------
extern "C" void kernel_launch(void* const* d_in, const int* in_sizes, int n_in,
                              void* d_out, int out_size, void* d_ws, size_t ws_size,
                              hipStream_t stream) {
  const float* means3D = (const float*)d_in[0];
  // d_in[1] = means2D (unused by reference math)
  const float* opac    = (const float*)d_in[2];
  const float* colors  = (const float*)d_in[3];
  const float* scales  = (const float*)d_in[4];
  const float* rots    = (const float*)d_in[5];
  const float* bg      = (const float*)d_in[6];
  const float* viewm   = (const float*)d_in[7];
  const float* projm   = (const float*)d_in[8];
  // d_in[9] = campos (unused)
  float* recs = (float*)d_ws;     // 1024 * 12 floats = 48 KB sorted records

  gs_prep_sort<<<1, 1024, 0, stream>>>(means3D, opac, colors, scales, rots,
                                       viewm, projm, recs);
  gs_raster<<<64, 256, 0, stream>>>(recs, bg, (float*)d_out);
}